// CentroidLoss_40870908788740
// MI455X (gfx1250) — compile-verified
//
#include <hip/hip_runtime.h>

#define CLS 64
#define DIM 128
#define EPSF 1e-6f
#define MARGIN 1.4f

typedef __attribute__((ext_vector_type(2))) float v2f;
typedef __attribute__((ext_vector_type(8))) float v8f;

// ---------------- workspace layout ----------------
// byte 0   : double  sumsq           (Σ ||e||^2)
// byte 256 : float   counts[64]
// byte 512 : float   sums[64*128]
// --------------------------------------------------

__global__ void zero_ws_kernel(float* __restrict__ sums, float* __restrict__ counts,
                               double* __restrict__ sumsq) {
    const int i = blockIdx.x * blockDim.x + threadIdx.x;
    if (i < CLS * DIM) sums[i] = 0.0f;
    if (i < CLS)       counts[i] = 0.0f;
    if (i == 0)        *sumsq = 0.0;
}

// One pass over the 128 MB of embeddings: per-class sums (LDS privatized),
// counts, and total sum of squares.
__global__ __launch_bounds__(256) void accum_kernel(
    const float* __restrict__ emb, const int* __restrict__ tgt,
    float* __restrict__ g_sums, float* __restrict__ g_counts,
    double* __restrict__ g_sumsq, int n_rows) {
    __shared__ float lsums[CLS * DIM];   // 32 KB
    __shared__ float lcounts[CLS];
    __shared__ float lsq[8];

    const int tid = threadIdx.x;
    for (int i = tid; i < CLS * DIM; i += 256) lsums[i] = 0.0f;
    if (tid < CLS) lcounts[tid] = 0.0f;
    __syncthreads();

    const int wave = tid >> 5;
    const int lane = tid & 31;
    const int gw = blockIdx.x * 8 + wave;          // global wave id
    const int wstride = gridDim.x * 8;

    float sq = 0.0f;
    for (int row = gw; row < n_rows; row += wstride) {
        const float4 v =
            *reinterpret_cast<const float4*>(emb + (size_t)row * DIM + lane * 4);
        const int t = tgt[row];
        sq += v.x * v.x + v.y * v.y + v.z * v.z + v.w * v.w;
        float* dst = &lsums[t * DIM + lane * 4];
        atomicAdd(dst + 0, v.x);                   // ds_add_f32
        atomicAdd(dst + 1, v.y);
        atomicAdd(dst + 2, v.z);
        atomicAdd(dst + 3, v.w);
        if (lane == 0) atomicAdd(&lcounts[t], 1.0f);
    }

    // wave32 reduction of sum-of-squares
    sq += __shfl_xor(sq, 16, 32);
    sq += __shfl_xor(sq, 8, 32);
    sq += __shfl_xor(sq, 4, 32);
    sq += __shfl_xor(sq, 2, 32);
    sq += __shfl_xor(sq, 1, 32);
    if (lane == 0) lsq[wave] = sq;
    __syncthreads();

    // flush block partials to global (native FP atomics)
    for (int i = tid; i < CLS * DIM; i += 256) {
        const float v = lsums[i];
        if (v != 0.0f) unsafeAtomicAdd(&g_sums[i], v);
    }
    if (tid < CLS) {
        const float v = lcounts[tid];
        if (v != 0.0f) unsafeAtomicAdd(&g_counts[tid], v);
    }
    if (tid == 0) {
        double s = 0.0;
        for (int w = 0; w < 8; ++w) s += (double)lsq[w];
        unsafeAtomicAdd(g_sumsq, s);
    }
}

// Single block: centers, 64x64 Gram via v_wmma_f32_16x16x4_f32, hinge loss,
// and the algebraic l2 term.
__global__ __launch_bounds__(256) void finalize_kernel(
    const float* __restrict__ g_sums, const float* __restrict__ g_counts,
    const double* __restrict__ g_sumsq, float* __restrict__ out, int n_rows) {
    __shared__ __attribute__((aligned(16))) float cen[CLS * DIM];  // 32 KB
    __shared__ float Q[CLS];
    __shared__ float red[CLS];
    __shared__ float wsum[8];

    const int tid = threadIdx.x;

    // centers = sums / (counts + eps)
    for (int i = tid; i < CLS * DIM; i += 256) {
        const int c = i >> 7;
        cen[i] = g_sums[i] / (g_counts[c] + EPSF);
    }
    __syncthreads();

    // per-class Q_c = ||cen_c||^2 and correction term counts*Q - 2*sums.cen
    if (tid < CLS) {
        float q = 0.0f, s = 0.0f;
        const float* cr = &cen[tid * DIM];
        const float* sr = &g_sums[tid * DIM];
        for (int d = 0; d < DIM; ++d) {
            q += cr[d] * cr[d];
            s += sr[d] * cr[d];
        }
        Q[tid] = q;
        red[tid] = g_counts[tid] * q - 2.0f * s;
    }
    __syncthreads();

    // Gram matrix G = Cen * Cen^T  (64x64 = 4x4 grid of 16x16 tiles),
    // 2 tiles per wave, K-loop of 32 x v_wmma_f32_16x16x4_f32.
    const int wave = tid >> 5;
    const int lane = tid & 31;
    const int half = lane >> 4;   // 0: lanes 0-15, 1: lanes 16-31
    const int l = lane & 15;

    float local = 0.0f;
#pragma unroll
    for (int ti = 0; ti < 2; ++ti) {
        const int tile = wave * 2 + ti;
        const int m0 = (tile >> 2) * 16;
        const int n0 = (tile & 3) * 16;
        const float* arow = &cen[(m0 + l) * DIM];
        const float* brow = &cen[(n0 + l) * DIM];
        v8f acc = {};
        for (int k0 = 0; k0 < DIM; k0 += 4) {
            // A 16x4 layout: lane(0-15) V0/V1 = K0/K1, lane(16-31) = K2/K3
            const v2f a = *reinterpret_cast<const v2f*>(arow + k0 + 2 * half);
            const v2f b = *reinterpret_cast<const v2f*>(brow + k0 + 2 * half);
            acc = __builtin_amdgcn_wmma_f32_16x16x4_f32(
                false, a, false, b, (short)0, acc, false, false);
        }
        // D layout: element r @ lane(half,l) = G[m0 + r + 8*half][n0 + l]
#pragma unroll
        for (int r = 0; r < 8; ++r) {
            const int i = m0 + r + 8 * half;
            const int j = n0 + l;
            if (j > i) {  // strict upper triangle, each pair exactly once
                const float d2 = Q[i] + Q[j] - 2.0f * acc[r];
                const float t = MARGIN - sqrtf(d2 + EPSF);
                if (t > 0.0f) local += t * t;
            }
        }
    }

    local += __shfl_xor(local, 16, 32);
    local += __shfl_xor(local, 8, 32);
    local += __shfl_xor(local, 4, 32);
    local += __shfl_xor(local, 2, 32);
    local += __shfl_xor(local, 1, 32);
    if (lane == 0) wsum[wave] = local;
    __syncthreads();

    if (tid == 0) {
        float ccsum = 0.0f;
        for (int w = 0; w < 8; ++w) ccsum += wsum[w];
        float corr = 0.0f;
        for (int c = 0; c < CLS; ++c) corr += red[c];
        const float l2 = (float)((*g_sumsq + (double)corr) / (double)n_rows);
        const float cc = ccsum / (float)CLS * (float)(CLS - 1) / 2.0f;
        out[0] = l2 + cc;
    }
}

extern "C" void kernel_launch(void* const* d_in, const int* in_sizes, int n_in,
                              void* d_out, int out_size, void* d_ws, size_t ws_size,
                              hipStream_t stream) {
    const float* emb = (const float*)d_in[0];
    const int* tgt = (const int*)d_in[1];
    float* out = (float*)d_out;
    const int n_rows = in_sizes[1];  // target count == N

    char* ws = (char*)d_ws;
    double* sumsq = (double*)(ws + 0);
    float* counts = (float*)(ws + 256);
    float* sums = (float*)(ws + 512);

    zero_ws_kernel<<<(CLS * DIM + 255) / 256, 256, 0, stream>>>(sums, counts, sumsq);
    accum_kernel<<<1024, 256, 0, stream>>>(emb, tgt, sums, counts, sumsq, n_rows);
    finalize_kernel<<<1, 256, 0, stream>>>(sums, counts, sumsq, out, n_rows);
}